// CTCLayer_50586124812451
// MI455X (gfx1250) — compile-verified
//
#include <hip/hip_runtime.h>
#include <cstdint>

// CTC loss (Keras ctc_batch_cost semantics), fused log-softmax + alpha scan.
// One workgroup per batch element: 160 threads = 5 wave32, one thread per
// extended-label state (S=129). Async global->LDS double buffering of each
// 512B probability row (gfx1250 global_load_async_to_lds_b32 / ASYNCcnt).
// 2 barriers per time step.

#define CT_T 256
#define CT_C 128
#define CT_L 64
#define CT_S 129          // 2*L+1
#define CT_BLANK 127      // C-1
#define CT_NEG (-1e30f)
#define CT_EPS (1e-7f)

#if __has_builtin(__builtin_amdgcn_s_wait_asynccnt)
#define ASYNC_WAIT(n) __builtin_amdgcn_s_wait_asynccnt(n)
#else
#define ASYNC_WAIT(n) asm volatile("s_wait_asynccnt %0" ::"n"(n) : "memory")
#endif

// LDS byte offset of a __shared__ object (generic -> addrspace(3) -> int).
__device__ __forceinline__ uint32_t lds_off(const void* p) {
  return (uint32_t)(uintptr_t)(__attribute__((address_space(3))) const void*)p;
}

// One 32-bit element per lane, global -> LDS, tracked by per-wave ASYNCcnt.
__device__ __forceinline__ void async_load_f32(uint32_t lds, const float* g) {
  asm volatile("global_load_async_to_lds_b32 %0, %1, off"
               :: "v"(lds), "v"(g)
               : "memory");
}

__device__ __forceinline__ float lae(float a, float b) {
  // logaddexp; exact identity lae(x, -1e30) == x in f32 (exp underflows to 0),
  // and lae(-1e30,-1e30) rounds back to -1e30, matching the reference scan.
  float m = fmaxf(a, b);
  float d = fabsf(a - b);
  return m + __logf(1.0f + __expf(-d));
}

__global__ __launch_bounds__(160) void ctc_loss_kernel(
    const int* __restrict__ y_true,   // [B, L]
    const float* __restrict__ y_pred, // [B, T, C]
    float* __restrict__ out)          // [B]
{
  __shared__ float praw[2][CT_C];     // double-buffered prob row (async dest)
  __shared__ float alphaA[CT_S + 2];  // [0..1] = log-zero halo
  __shared__ float alphaB[CT_S + 2];
  __shared__ float red[4];            // per-wave softmax-sum partials
  __shared__ int   extsh[CT_S];

  const int tid = (int)threadIdx.x;
  const int b   = (int)blockIdx.x;
  const float* yp = y_pred + (size_t)b * CT_T * CT_C;

  // Extended labels: blank, l1, blank, l2, ..., blank. Halo = log-zero.
  if (tid < CT_S) {
    extsh[tid] = (tid & 1) ? y_true[b * CT_L + (tid >> 1)] : CT_BLANK;
  }
  if (tid < 2) { alphaA[tid] = CT_NEG; alphaB[tid] = CT_NEG; }

  // Prefetch rows 0 and 1; each wave owns 32 contiguous floats and later
  // waits only on its OWN per-wave ASYNCcnt before the publishing barrier.
  if (tid < CT_C) {
    async_load_f32(lds_off(&praw[0][tid]), yp + tid);
    async_load_f32(lds_off(&praw[1][tid]), yp + CT_C + tid);
  }
  __syncthreads();  // extsh visible

  int es = CT_BLANK, sk = 0;
  if (tid < CT_S) {
    es = extsh[tid];
    int em2 = (tid >= 2) ? extsh[tid - 2] : CT_BLANK;
    sk = (tid >= 2) && (es != CT_BLANK) && (es != em2);
  }

  float* acur  = alphaA;
  float* anext = alphaB;

  // ---------------- t = 0: alpha init ----------------
  if (tid < CT_C) {
    ASYNC_WAIT(1);                              // row 0 resident (this wave)
    float s4 = praw[0][tid] + CT_EPS;
    #pragma unroll
    for (int off = 16; off > 0; off >>= 1) s4 += __shfl_xor(s4, off, 32);
    if ((tid & 31) == 0) red[tid >> 5] = s4;
  }
  __syncthreads();                              // praw[0] + red published
  if (tid < CT_S) {
    float lse = __logf(red[0] + red[1] + red[2] + red[3]);
    float na  = CT_NEG;
    if (tid < 2) na = __logf(praw[0][es] + CT_EPS) - lse;
    acur[2 + tid] = na;
  }
  __syncthreads();                              // gathers from praw[0] done
  if (tid < CT_C)                               // refill buffer 0 with row 2
    async_load_f32(lds_off(&praw[0][tid]), yp + 2 * CT_C + tid);

  // ---------------- t = 1 .. T-1: recursion ----------------
  for (int t = 1; t < CT_T; ++t) {
    const int buf = t & 1;

    if (tid < CT_C) {
      if (t < CT_T - 1) { ASYNC_WAIT(1); } else { ASYNC_WAIT(0); }
      float s4 = praw[buf][tid] + CT_EPS;
      #pragma unroll
      for (int off = 16; off > 0; off >>= 1) s4 += __shfl_xor(s4, off, 32);
      if ((tid & 31) == 0) red[tid >> 5] = s4;
    }
    __syncthreads();                            // row t + partials published

    if (tid < CT_S) {
      float lse = __logf(red[0] + red[1] + red[2] + red[3]);
      float lpe = __logf(praw[buf][es] + CT_EPS) - lse;  // direct gather
      float a0  = acur[2 + tid];
      float a1  = acur[1 + tid];
      float a2  = sk ? acur[tid] : CT_NEG;      // branchless skip transition
      anext[2 + tid] = lae(lae(a0, a1), a2) + lpe;
    }
    __syncthreads();                            // gathers + alpha writes done

    if ((t + 2 < CT_T) && tid < CT_C)           // refill consumed buffer
      async_load_f32(lds_off(&praw[buf][tid]), yp + (t + 2) * CT_C + tid);

    float* tswap = acur; acur = anext; anext = tswap;
  }

  // Last in-loop barrier ordered the final alpha writes before this read.
  if (tid == 0) {
    out[b] = -lae(acur[2 + CT_S - 1], acur[2 + CT_S - 2]);
  }
}

extern "C" void kernel_launch(void* const* d_in, const int* in_sizes, int n_in,
                              void* d_out, int out_size, void* d_ws, size_t ws_size,
                              hipStream_t stream) {
  const int*   y_true = (const int*)d_in[0];   // [B, L] int32
  const float* y_pred = (const float*)d_in[1]; // [B, T, C] float32
  float*       out    = (float*)d_out;         // [B, 1] float32

  const int B = in_sizes[0] / CT_L;
  hipLaunchKernelGGL(ctc_loss_kernel, dim3(B), dim3(160), 0, stream,
                     y_true, y_pred, out);
}